// HungarianMatcher3D_69947837382907
// MI455X (gfx1250) — compile-verified
//
#include <hip/hip_runtime.h>
#include <hip/hip_bf16.h>
#include <math.h>

typedef __attribute__((ext_vector_type(2))) float v2f;
typedef __attribute__((ext_vector_type(8))) float v8f;
typedef __attribute__((ext_vector_type(4))) unsigned int u32x4;
typedef __attribute__((ext_vector_type(8))) int i32x8;
typedef __attribute__((ext_vector_type(4))) int i32x4;
typedef __attribute__((address_space(3))) char lds_char;

#define HM_EPS 1e-6f
#define NC 256
#define AP 260  // LDS pitch for A tile: 256 + 4 floats -> conflict-free ds_load_b64

#ifndef HM_USE_TDM
#define HM_USE_TDM 1
#endif

// ---------------------------------------------------------------------------
// Kernel 1: row softmax over nc=256. One wave32 per row, 8 values per lane.
// ---------------------------------------------------------------------------
__global__ void __launch_bounds__(256)
hm3d_softmax(const float* __restrict__ logits, float* __restrict__ probs, int nrows) {
  int wave = (int)((blockIdx.x * 256u + threadIdx.x) >> 5);
  int lane = threadIdx.x & 31;
  if (wave >= nrows) return;
  const float* row = logits + (size_t)wave * NC;
  float v[8];
  float mx = -INFINITY;
#pragma unroll
  for (int j = 0; j < 8; ++j) { v[j] = row[lane + 32 * j]; mx = fmaxf(mx, v[j]); }
#pragma unroll
  for (int o = 16; o > 0; o >>= 1) mx = fmaxf(mx, __shfl_xor(mx, o, 32));
  float s = 0.0f;
#pragma unroll
  for (int j = 0; j < 8; ++j) { v[j] = __expf(v[j] - mx); s += v[j]; }
#pragma unroll
  for (int o = 16; o > 0; o >>= 1) s += __shfl_xor(s, o, 32);
  float inv = 1.0f / s;
  float* orow = probs + (size_t)wave * NC;
#pragma unroll
  for (int j = 0; j < 8; ++j) orow[lane + 32 * j] = v[j] * inv;
}

// ---------------------------------------------------------------------------
// Kernel 2: per-box stats: [0..5]=box, [6..8]=corner min, [9..11]=corner max,
// [12]=volume, padded to stride 16 floats.
// ---------------------------------------------------------------------------
__global__ void __launch_bounds__(256)
hm3d_stats(const float* __restrict__ boxes, const float* __restrict__ corners,
           float* __restrict__ stats, int n) {
  int idx = blockIdx.x * 256 + threadIdx.x;
  if (idx >= n) return;
  const float* c = corners + (size_t)idx * 24;
  float mn[3] = {c[0], c[1], c[2]};
  float mx[3] = {c[0], c[1], c[2]};
#pragma unroll
  for (int p = 1; p < 8; ++p)
#pragma unroll
    for (int d = 0; d < 3; ++d) {
      float v = c[p * 3 + d];
      mn[d] = fminf(mn[d], v);
      mx[d] = fmaxf(mx[d], v);
    }
  float vol = (mx[0] - mn[0]) * (mx[1] - mn[1]) * (mx[2] - mn[2]);
  float* s = stats + (size_t)idx * 16;
#pragma unroll
  for (int k = 0; k < 6; ++k) s[k] = boxes[(size_t)idx * 6 + k];
#pragma unroll
  for (int d = 0; d < 3; ++d) { s[6 + d] = mn[d]; s[9 + d] = mx[d]; }
  s[12] = vol;
}

// ---------------------------------------------------------------------------
// Kernel 3: main cost kernel. 8 waves/block; each wave owns one 16x16 tile.
// The 16x256 probs slab (shared by all 8 waves) is staged once into LDS by the
// Tensor Data Mover (pad_interval=256 DW, pad_amount=4 DW -> pitch 260), then
// the K loop runs WMMA f32 16x16x4 against an on-the-fly one-hot B fragment
// (exact gather). Target stats + pred stats also live in LDS.
// ---------------------------------------------------------------------------
__global__ void __launch_bounds__(256)
hm3d_cost(const float* __restrict__ probs, const float* __restrict__ pstats,
          const float* __restrict__ tstats, const int* __restrict__ labels,
          float* __restrict__ out, int Mtot, int blocksPerRow) {
  __shared__ float s_A[16 * AP];     // A tile: 16 rows x 256 (pitch 260)
  __shared__ float s_t[128 * 17];    // target stats, padded pitch
  __shared__ float s_p[16 * 16];     // pred stats for the 16 rows
  __shared__ int s_lab[128];

  const int tid = threadIdx.x;
  const int lane = tid & 31;
  const int waveId = tid >> 5;
  const int itile = blockIdx.x / blocksPerRow;         // row tile (16 preds)
  const int bcol = blockIdx.x - itile * blocksPerRow;  // block column
  const int m0 = bcol * 128;                           // first target col
  const int n0 = itile * 16;

#if HM_USE_TDM
  // ---- TDM: DMA the 16x256 f32 probs tile into LDS with padded pitch ----
  if (waveId == 0) {
    unsigned long long ga =
        (unsigned long long)(const void*)probs + (unsigned long long)n0 * NC * 4ull;
    unsigned lds_off = (unsigned)(unsigned long long)(lds_char*)(&s_A[0]);
    u32x4 g0;
    g0.x = 0x1u;                                        // count=1, user D#
    g0.y = lds_off;                                     // lds_addr (bytes)
    g0.z = (unsigned)ga;                                // global_addr[31:0]
    g0.w = (unsigned)((ga >> 32) & 0x1FFFFFFull) | (2u << 30);  // addr[56:32]|type=2
    i32x8 g1;
    g1[0] = 0x07D20000;        // data_size=4B, pad_enable, interval=256DW, amount=4DW
    g1[1] = (int)(256u << 16); // tensor_dim0 = 256 (low 16 bits at [63:48])
    g1[2] = (int)(16u << 16);  // tensor_dim0 hi=0 | tensor_dim1 = 16
    g1[3] = (int)(256u << 16); // tensor_dim1 hi=0 | tile_dim0 = 256
    g1[4] = 16;                // tile_dim1 = 16, tile_dim2 = 0
    g1[5] = 256;               // tensor_dim0_stride = 256 (elements)
    g1[6] = 0;                 // stride0 hi | stride1 lo
    g1[7] = 0;
    i32x4 g2 = {};             // 2D tensor: groups 2/3 zero
    i32x4 g3 = {};
#if __clang_major__ >= 23
    i32x8 g4 = {};
    __builtin_amdgcn_tensor_load_to_lds(g0, g1, g2, g3, g4, 0);
#else
    __builtin_amdgcn_tensor_load_to_lds(g0, g1, g2, g3, 0);
#endif
  }
#else
  // ---- Fallback: cooperative float4 staging of the A tile ----
  {
    const float* gA = probs + (size_t)n0 * NC;
    for (int i4 = tid; i4 < (16 * NC) / 4; i4 += 256) {
      int idx = i4 * 4;
      int row = idx >> 8;
      int k = idx & (NC - 1);
      *(float4*)(&s_A[row * AP + k]) = *(const float4*)(gA + idx);
    }
  }
#endif

  // Stage 128 targets' stats + labels and the 16 pred-stat rows into LDS.
  const float* tg = tstats + (size_t)m0 * 16;
  for (int idx = tid; idx < 128 * 16; idx += 256) {
    int c = idx >> 4, k = idx & 15;
    s_t[c * 17 + k] = tg[idx];
  }
  if (tid < 128) s_lab[tid] = labels[m0 + tid];
  s_p[tid] = pstats[(size_t)n0 * 16 + tid];

#if HM_USE_TDM
  if (waveId == 0) __builtin_amdgcn_s_wait_tensorcnt(0);
#endif
  __syncthreads();

  // A fragment addressing: 16x4 f32, lanes 0-15 hold K=0,1; lanes 16-31 K=2,3.
  const int rowA = lane & 15;
  const int koff = (lane >> 4) << 1;
  const float* aRow = &s_A[rowA * AP + koff];

  const int lcol = (waveId << 4) + (lane & 15);  // this wave's target column
  const int lab = s_lab[lcol];

  // C = P @ OneHot(labels): after the K loop acc[r] == out_prob[n, label[m]].
  v8f acc = {};
#pragma unroll 8
  for (int kb = 0; kb < NC; kb += 4) {
    v2f a = *(const v2f*)(aRow + kb);  // ds_load_b64, conflict-free (pitch 260)
    v2f b;
    b.x = (lab == kb + koff) ? 1.0f : 0.0f;
    b.y = (lab == kb + koff + 1) ? 1.0f : 0.0f;
    acc = __builtin_amdgcn_wmma_f32_16x16x4_f32(
        /*neg_a=*/false, a, /*neg_b=*/false, b,
        /*c_mod=*/(short)0, acc, /*reuse_a=*/false, /*reuse_b=*/false);
  }

  // Target-side scalars for this column (from LDS).
  const float* tcol = &s_t[lcol * 17];
  const float tb0 = tcol[0], tb1 = tcol[1], tb2 = tcol[2];
  const float tb3 = tcol[3], tb4 = tcol[4], tb5 = tcol[5];
  const float tmn0 = tcol[6], tmn1 = tcol[7], tmn2 = tcol[8];
  const float tmx0 = tcol[9], tmx1 = tcol[10], tmx2 = tcol[11];
  const float tvol = tcol[12];

  const int m = m0 + lcol;
  const int rbase = (lane >> 4) << 3;  // lanes 16-31 hold M=r+8

#pragma unroll
  for (int r = 0; r < 8; ++r) {
    const int n = n0 + rbase + r;
    const float* ps = &s_p[(rbase + r) * 16];  // LDS broadcast within half-wave
    // L1 box distance
    float cb = fabsf(ps[0] - tb0) + fabsf(ps[1] - tb1) + fabsf(ps[2] - tb2) +
               fabsf(ps[3] - tb3) + fabsf(ps[4] - tb4) + fabsf(ps[5] - tb5);
    // axis-aligned 3D GIoU
    const float pmn0 = ps[6], pmn1 = ps[7], pmn2 = ps[8];
    const float pmx0 = ps[9], pmx1 = ps[10], pmx2 = ps[11];
    const float pvol = ps[12];
    float i0 = fminf(pmx0, tmx0) - fmaxf(pmn0, tmn0);
    float i1 = fminf(pmx1, tmx1) - fmaxf(pmn1, tmn1);
    float i2 = fminf(pmx2, tmx2) - fmaxf(pmn2, tmn2);
    float inter = fmaxf(i0, 0.0f) * fmaxf(i1, 0.0f) * fmaxf(i2, 0.0f);
    float e0 = fmaxf(pmx0, tmx0) - fminf(pmn0, tmn0);
    float e1 = fmaxf(pmx1, tmx1) - fminf(pmn1, tmn1);
    float e2 = fmaxf(pmx2, tmx2) - fminf(pmn2, tmn2);
    float enclose = e0 * e1 * e2;
    float uni = pvol + tvol - inter;
    float giou = inter / (uni + HM_EPS) - (enclose - uni) / (enclose + HM_EPS);
    // C = cost_bbox + cost_class - giou ; cost_class = -prob
    out[(size_t)n * Mtot + m] = cb - acc[r] - giou;
  }
}

// ---------------------------------------------------------------------------
// Launch
// ---------------------------------------------------------------------------
extern "C" void kernel_launch(void* const* d_in, const int* in_sizes, int n_in,
                              void* d_out, int out_size, void* d_ws, size_t ws_size,
                              hipStream_t stream) {
  const float* logits   = (const float*)d_in[0];  // [bs, nq, 256]
  const float* pboxes   = (const float*)d_in[1];  // [bs, nq, 6]
  const float* pcorners = (const float*)d_in[2];  // [bs, nq, 8, 3]
  const int*   tlabels  = (const int*)d_in[3];    // [M]
  const float* tboxes   = (const float*)d_in[4];  // [M, 6]
  const float* tcorners = (const float*)d_in[5];  // [M, 8, 3]
  float* out = (float*)d_out;                     // [bs*nq, M]

  const int N = in_sizes[0] / NC;  // 16384 flattened predictions
  const int M = in_sizes[3];       // 1024 targets

  // Workspace layout: probs [N*256], pstats [N*16], tstats [M*16] (floats).
  float* probs  = (float*)d_ws;
  float* pstats = probs + (size_t)N * NC;
  float* tstats = pstats + (size_t)N * 16;

  hm3d_softmax<<<(N * 32 + 255) / 256, 256, 0, stream>>>(logits, probs, N);
  hm3d_stats<<<(N + 255) / 256, 256, 0, stream>>>(pboxes, pcorners, pstats, N);
  hm3d_stats<<<(M + 255) / 256, 256, 0, stream>>>(tboxes, tcorners, tstats, M);

  const int blocksPerRow = M / 128;             // 8 col-tiles per block (1/wave)
  const int grid = (N / 16) * blocksPerRow;     // exact tiling, EXEC stays full
  hm3d_cost<<<grid, 256, 0, stream>>>(probs, pstats, tstats, tlabels, out, M, blocksPerRow);
}